// Model1_41034117546432
// MI455X (gfx1250) — compile-verified
//
#include <hip/hip_runtime.h>
#include <math.h>

#define T_LEN   1024
#define D_DIM   300
#define L_CH    16
#define CD_DIM  16
#define H_DIM   150
#define DIN     450     // D + 3*F
#define G4      600     // 4*H
#define FH_DIM  150
#define E_SP    512
#define MAXSPAN 10
#define NFE     50
#define SPAN_DIM 1000   // 3*D + 2*FE
#define A_ACT   2000
#define NROWS   (E_SP * MAXSPAN)   // 5120 span-slot rows

typedef float v2f __attribute__((ext_vector_type(2)));
typedef float v8f __attribute__((ext_vector_type(8)));

__device__ __forceinline__ float sigf(float x) { return 1.f / (1.f + expf(-x)); }

// ---------------- Generic NT GEMM via f32 WMMA 16x16x4 ----------------
// C[m,n] = sum_k A[m*lda+k] * B[n*ldb+k]  (+ bias[n]) (relu optional)
// One wave per 16x16 C tile. K tail (K%4) handled with scalar FMAs into the
// accumulator fragment. Ragged M/N handled by clamped fragment rows + masked
// store (WMMA itself always runs with EXEC all-1s; the tile guard is
// wave-uniform).
// Fragment layout (ISA 7.12.2): A 16x4: lane%16 = M, lane/16 selects K pair
// {0,1}/{2,3}; B 4x16 mirrors with lane%16 = N; C: VGPR r -> M = r + 8*(lane/16).
__global__ void k_gemm_nt_wmma(const float* __restrict__ A, const float* __restrict__ B,
                               const float* __restrict__ bias, float* __restrict__ C,
                               int M, int N, int K, int lda, int ldb, int ldc, int relu)
{
    int ntiles = (N + 15) >> 4;
    int mtiles = (M + 15) >> 4;
    int wave = (blockIdx.x * blockDim.x + threadIdx.x) >> 5;
    if (wave >= mtiles * ntiles) return;            // wave-uniform
    int lane = threadIdx.x & 31;
    int mt = wave / ntiles, nt = wave - mt * ntiles;
    int m0 = mt * 16, n0 = nt * 16;
    int lm = lane & 15;
    int kh = (lane >> 4) << 1;                      // 0 or 2
    int am = m0 + lm; if (am > M - 1) am = M - 1;   // clamp (edge tiles)
    int bn = n0 + lm; if (bn > N - 1) bn = N - 1;
    const float* arow = A + (long)am * lda;
    const float* brow = B + (long)bn * ldb;
    v8f acc = {0.f, 0.f, 0.f, 0.f, 0.f, 0.f, 0.f, 0.f};
    int kmain = K & ~3;
    for (int k0 = 0; k0 < kmain; k0 += 4) {
        v2f a, b;
        a.x = arow[k0 + kh];
        a.y = arow[k0 + kh + 1];
        b.x = brow[k0 + kh];
        b.y = brow[k0 + kh + 1];
        acc = __builtin_amdgcn_wmma_f32_16x16x4_f32(false, a, false, b,
                                                    (short)0, acc, false, false);
    }
    int mbase = m0 + ((lane >> 4) << 3);            // +8 for upper half-lanes
    // scalar K tail: lane owns C elements (mbase+r, n0+lm)
    for (int k = kmain; k < K; ++k) {
        float bv = brow[k];
        #pragma unroll
        for (int r = 0; r < 8; ++r) {
            int ar = mbase + r; if (ar > M - 1) ar = M - 1;
            acc[r] += A[(long)ar * lda + k] * bv;
        }
    }
    float badd = bias ? bias[bn] : 0.f;
    int col = n0 + lm;
    if (col < N) {
        #pragma unroll
        for (int r = 0; r < 8; ++r) {
            int row = mbase + r;
            if (row < M) {
                float v = acc[r] + badd;
                if (relu) v = fmaxf(v, 0.f);
                C[(long)row * ldc + col] = v;
            }
        }
    }
}

// ---------------- transpose: dst[c*R + r] = src[r*Cc + c] ----------------
__global__ void k_transpose(const float* __restrict__ src, float* __restrict__ dst,
                            int R, int Cc)
{
    int idx = blockIdx.x * blockDim.x + threadIdx.x;
    if (idx >= R * Cc) return;
    int r = idx / Cc, c = idx - r * Cc;
    dst[c * R + r] = src[idx];
}

// ---------------- K0: embedding gather + char CNN ----------------
__global__ void k_embed_char(const float* __restrict__ fast_text,
                             const float* __restrict__ char_table,
                             const float* __restrict__ w3, const float* __restrict__ b3,
                             const float* __restrict__ w4, const float* __restrict__ b4,
                             const float* __restrict__ w5, const float* __restrict__ b5,
                             const int* __restrict__ word_ids,
                             const int* __restrict__ char_ids,
                             float* __restrict__ ft, float* __restrict__ lstm_in)
{
    __shared__ float ce[16][17];           // [pos][chan]
    int t = blockIdx.x, tid = threadIdx.x;
    if (tid < 256) {
        int p = tid >> 4, c = tid & 15;
        ce[p][c] = char_table[char_ids[t * L_CH + p] * CD_DIM + c];
    }
    __syncthreads();
    if (tid < 150) {
        int f, K; const float *w, *bb;
        if (tid < 50)       { f = tid;       K = 3; w = w3; bb = b3; }
        else if (tid < 100) { f = tid - 50;  K = 4; w = w4; bb = b4; }
        else                { f = tid - 100; K = 5; w = w5; bb = b5; }
        const float* wf = w + f * CD_DIM * K;
        float best = 0.f;                  // max over relu(y) >= 0
        for (int p = 0; p <= L_CH - K; ++p) {
            float acc = bb[f];
            for (int c = 0; c < CD_DIM; ++c)
                for (int j = 0; j < K; ++j)
                    acc += wf[c * K + j] * ce[p + j][c];
            best = fmaxf(best, fmaxf(acc, 0.f));
        }
        lstm_in[t * DIN + D_DIM + tid] = best;
    }
    int wid = word_ids[t];
    for (int i = tid; i < D_DIM; i += blockDim.x) {
        float v = fast_text[wid * D_DIM + i];
        ft[t * D_DIM + i] = v;
        lstm_in[t * DIN + i] = v;
    }
}

// ---------------- K2: sequential BiLSTM (one block per direction) ----------------
__global__ void k_lstm(const float* __restrict__ Xi_f, const float* __restrict__ Xi_b,
                       const float* __restrict__ Wh_f, const float* __restrict__ Wh_b,
                       float* __restrict__ keys)
{
    int dir = blockIdx.x;
    const float* Xi = dir ? Xi_b : Xi_f;
    const float* Wh = dir ? Wh_b : Wh_f;
    __shared__ float h_s[H_DIM], c_s[H_DIM], z_s[G4];
    int tid = threadIdx.x;
    if (tid < H_DIM) { h_s[tid] = 0.f; c_s[tid] = 0.f; }
    __syncthreads();
    for (int step = 0; step < T_LEN; ++step) {
        int t = dir ? (T_LEN - 1 - step) : step;
        if (tid < G4) {
            float acc = Xi[t * G4 + tid];
            const float* wr = Wh + tid * H_DIM;
            for (int k = 0; k < H_DIM; ++k) acc += wr[k] * h_s[k];
            z_s[tid] = acc;
        }
        __syncthreads();
        if (tid < H_DIM) {
            float ig = sigf(z_s[tid]);
            float fg = sigf(z_s[H_DIM + tid]);
            float gg = tanhf(z_s[2 * H_DIM + tid]);
            float og = sigf(z_s[3 * H_DIM + tid]);
            float c  = fg * c_s[tid] + ig * gg;
            c_s[tid] = c;
            float h = og * tanhf(c);
            h_s[tid] = h;
            keys[t * D_DIM + dir * H_DIM + tid] = h;
        }
        __syncthreads();
    }
}

// ---------------- gather k_sp rows: ksp[(e*10+slot), :] = keys[clip(s+slot)] ----
__global__ void k_span_gather(const float* __restrict__ keys,
                              const int* __restrict__ espans, float* __restrict__ ksp)
{
    int idx = blockIdx.x * blockDim.x + threadIdx.x;
    if (idx >= NROWS * D_DIM) return;
    int r = idx / D_DIM, i = idx - r * D_DIM;
    int e = r / MAXSPAN, slot = r - e * MAXSPAN;
    int t = espans[e * 4] + slot; if (t > T_LEN - 1) t = T_LEN - 1;
    ksp[idx] = keys[t * D_DIM + i];
}

// ---------------- logits[r] = mask ? dot(h2[r], wout) : -1e30 ----------------
__global__ void k_logits(const float* __restrict__ h2, const float* __restrict__ wout,
                         const int* __restrict__ espans, float* __restrict__ logits)
{
    int r = blockIdx.x * blockDim.x + threadIdx.x;
    if (r >= NROWS) return;
    int e = r / MAXSPAN, slot = r - e * MAXSPAN;
    int s = espans[e * 4], eend = espans[e * 4 + 1];
    float acc = 0.f;
    const float* hr = h2 + r * FH_DIM;
    for (int k = 0; k < FH_DIM; ++k) acc += hr[k] * wout[k];
    logits[r] = (slot < (eend - s)) ? acc : -1e30f;
}

// ---------------- softmax + attention-pooled value + build enc row ----------------
__global__ void k_span_enc(const float* __restrict__ logits, const float* __restrict__ keys,
                           const float* __restrict__ ft, const float* __restrict__ feat_table,
                           const int* __restrict__ espans, float* __restrict__ enc)
{
    int e = blockIdx.x, tid = threadIdx.x;
    __shared__ float attn[MAXSPAN];
    __shared__ float fscale[2];
    int s = espans[e * 4 + 0], eend = espans[e * 4 + 1];
    if (tid == 0) {
        float mx = -1e30f;
        for (int i = 0; i < MAXSPAN; ++i) mx = fmaxf(mx, logits[e * MAXSPAN + i]);
        float sum = 0.f;
        for (int i = 0; i < MAXSPAN; ++i) { float v = expf(logits[e * MAXSPAN + i] - mx); attn[i] = v; sum += v; }
        float inv = 1.f / sum;
        for (int i = 0; i < MAXSPAN; ++i) attn[i] *= inv;
    }
    if (tid == 1 || tid == 2) {
        int fi = espans[e * 4 + 1 + tid];   // tid=1 -> feat0, tid=2 -> feat1
        float ss = 0.f;
        for (int d = 0; d < NFE; ++d) { float v = feat_table[fi * NFE + d]; ss += v * v; }
        fscale[tid - 1] = fminf(1.f, 1.f / fmaxf(sqrtf(ss), 1e-7f));
    }
    __syncthreads();
    float* erow = enc + e * SPAN_DIM;
    for (int i = tid; i < D_DIM; i += blockDim.x) {
        erow[i]           = keys[s * D_DIM + i];
        erow[D_DIM + i]   = keys[(eend - 1) * D_DIM + i];
        float acc = 0.f;
        for (int sl = 0; sl < MAXSPAN; ++sl) {
            int idx = s + sl; if (idx > T_LEN - 1) idx = T_LEN - 1;
            acc += attn[sl] * ft[idx * D_DIM + i];
        }
        erow[2 * D_DIM + i] = acc;
    }
    for (int i = tid; i < 2 * NFE; i += blockDim.x) {
        int j = i / NFE, d = i - j * NFE;
        int fi = espans[e * 4 + 2 + j];
        erow[3 * D_DIM + i] = feat_table[fi * NFE + d] * fscale[j];
    }
}

// ---------------- renorm actor rows ----------------
__global__ void k_renorm_actors(const float* __restrict__ actors, float* __restrict__ actors_n)
{
    int r = blockIdx.x * blockDim.x + threadIdx.x;
    if (r >= A_ACT) return;
    const float* a = actors + r * D_DIM;
    float ss = 0.f;
    for (int k = 0; k < D_DIM; ++k) ss += a[k] * a[k];
    float sc = fminf(1.f, 1.f / fmaxf(sqrtf(ss), 1e-7f));
    float* o = actors_n + r * D_DIM;
    for (int k = 0; k < D_DIM; ++k) o[k] = a[k] * sc;
}

// ---------------- score_1 = logsumexp(score_link, axis=1) ----------------
__global__ void k_lse(const float* __restrict__ sl, float* __restrict__ out)
{
    int row = blockIdx.x, tid = threadIdx.x;
    __shared__ float red[256];
    float mx = -1e30f;
    for (int i = tid; i < 512; i += 256) mx = fmaxf(mx, sl[row * 512 + i]);
    red[tid] = mx; __syncthreads();
    for (int s = 128; s > 0; s >>= 1) { if (tid < s) red[tid] = fmaxf(red[tid], red[tid + s]); __syncthreads(); }
    float m = red[0]; __syncthreads();
    float sum = 0.f;
    for (int i = tid; i < 512; i += 256) sum += expf(sl[row * 512 + i] - m);
    red[tid] = sum; __syncthreads();
    for (int s = 128; s > 0; s >>= 1) { if (tid < s) red[tid] += red[tid + s]; __syncthreads(); }
    if (tid == 0) out[row] = m + logf(red[0]);
}

static inline int gemm_blocks(int M, int N)
{
    int waves = ((M + 15) / 16) * ((N + 15) / 16);
    return (waves * 32 + 255) / 256;
}

extern "C" void kernel_launch(void* const* d_in, const int* in_sizes, int n_in,
                              void* d_out, int out_size, void* d_ws, size_t ws_size,
                              hipStream_t stream)
{
    const float* fast_text   = (const float*)d_in[0];
    const float* actor_mat   = (const float*)d_in[1];
    const float* char_table  = (const float*)d_in[2];
    const float* conv_w3     = (const float*)d_in[3];
    const float* conv_b3     = (const float*)d_in[4];
    const float* conv_w4     = (const float*)d_in[5];
    const float* conv_b4     = (const float*)d_in[6];
    const float* conv_w5     = (const float*)d_in[7];
    const float* conv_b5     = (const float*)d_in[8];
    const float* Wi_f        = (const float*)d_in[9];
    const float* Wh_f        = (const float*)d_in[10];
    const float* b_f         = (const float*)d_in[11];
    const float* Wi_b        = (const float*)d_in[12];
    const float* Wh_b        = (const float*)d_in[13];
    const float* b_b         = (const float*)d_in[14];
    const float* alpha_W1    = (const float*)d_in[15];
    const float* alpha_b1    = (const float*)d_in[16];
    const float* alpha_W2    = (const float*)d_in[17];
    const float* alpha_b2    = (const float*)d_in[18];
    const float* alpha_wout  = (const float*)d_in[19];
    const float* feat_table  = (const float*)d_in[20];
    const float* affine_W    = (const float*)d_in[21];
    const float* affine_b    = (const float*)d_in[22];
    const int*   word_ids    = (const int*)d_in[23];
    const int*   char_ids    = (const int*)d_in[24];
    const int*   espans      = (const int*)d_in[25];

    // ---- workspace layout (floats), with liveness-based reuse ----
    float* ws = (float*)d_ws;
    float* ft       = ws;                       // 307200      [K0 .. span_enc]
    float* regionA  = ws + 307200;              // 1689600
    float*   lstm_in = regionA;                 //   460800    [K0 .. gates]
    float*   Xi_f    = regionA + 460800;        //   614400    [gates .. lstm]
    float*   Xi_b    = regionA + 1075200;       //   614400    [gates .. lstm]
    float*   ksp     = regionA;                 //   1536000   [gather .. h1 gemm] (reuse)
    float* keys     = ws + 1996800;             // 307200      [lstm .. span_enc]
    float* logits   = ws + 2304000;             // 5120
    float* regionB  = ws + 2309120;             // 1536000
    float*   h1      = regionB;                 //   768000    [h1 gemm .. h2 gemm]
    float*   h2      = regionB + 768000;        //   768000    [h2 gemm .. logits]
    float*   enc     = regionB;                 //   512000    [span_enc .. affine] (reuse)
    float*   small_  = regionB + 512000;        //   153600    [affine .. score]    (reuse)
    float* actors_n = ws + 3845120;             // 600000
    float* W1T      = ws + 4445120;             // 45000  (150x300)
    float* W2T      = ws + 4490120;             // 22500  (150x150)
    float* WAT      = ws + 4512620;             // 300000 (300x1000)
    // total 4,812,620 floats ~= 18.4 MiB

    float* out        = (float*)d_out;
    float* score_1    = out;                    // 2000
    float* score_link = out + A_ACT;            // 2000*512

    // weight transposes into NT form (K-contiguous rows)
    k_transpose<<<(D_DIM * FH_DIM + 255) / 256, 256, 0, stream>>>(alpha_W1, W1T, D_DIM, FH_DIM);
    k_transpose<<<(FH_DIM * FH_DIM + 255) / 256, 256, 0, stream>>>(alpha_W2, W2T, FH_DIM, FH_DIM);
    k_transpose<<<(SPAN_DIM * D_DIM + 255) / 256, 256, 0, stream>>>(affine_W, WAT, SPAN_DIM, D_DIM);

    k_embed_char<<<T_LEN, 256, 0, stream>>>(fast_text, char_table,
        conv_w3, conv_b3, conv_w4, conv_b4, conv_w5, conv_b5,
        word_ids, char_ids, ft, lstm_in);

    // Xi = lstm_in @ Wi^T + b   (M=1024, N=600, K=450)  — WMMA with scalar K-tail
    k_gemm_nt_wmma<<<gemm_blocks(T_LEN, G4), 256, 0, stream>>>(
        lstm_in, Wi_f, b_f, Xi_f, T_LEN, G4, DIN, DIN, DIN, G4, 0);
    k_gemm_nt_wmma<<<gemm_blocks(T_LEN, G4), 256, 0, stream>>>(
        lstm_in, Wi_b, b_b, Xi_b, T_LEN, G4, DIN, DIN, DIN, G4, 0);

    k_lstm<<<2, 640, 0, stream>>>(Xi_f, Xi_b, Wh_f, Wh_b, keys);

    // span attention MLP on matrix pipes
    k_span_gather<<<(NROWS * D_DIM + 255) / 256, 256, 0, stream>>>(keys, espans, ksp);
    // h1 = relu(ksp @ W1 + b1)   (M=5120, N=150, K=300)
    k_gemm_nt_wmma<<<gemm_blocks(NROWS, FH_DIM), 256, 0, stream>>>(
        ksp, W1T, alpha_b1, h1, NROWS, FH_DIM, D_DIM, D_DIM, D_DIM, FH_DIM, 1);
    // h2 = relu(h1 @ W2 + b2)    (M=5120, N=150, K=150)
    k_gemm_nt_wmma<<<gemm_blocks(NROWS, FH_DIM), 256, 0, stream>>>(
        h1, W2T, alpha_b2, h2, NROWS, FH_DIM, FH_DIM, FH_DIM, FH_DIM, FH_DIM, 1);
    k_logits<<<(NROWS + 255) / 256, 256, 0, stream>>>(h2, alpha_wout, espans, logits);

    k_span_enc<<<E_SP, 256, 0, stream>>>(logits, keys, ft, feat_table, espans, enc);

    // small = enc @ affine_W + affine_b   (M=512, N=300, K=1000)
    k_gemm_nt_wmma<<<gemm_blocks(E_SP, D_DIM), 256, 0, stream>>>(
        enc, WAT, affine_b, small_, E_SP, D_DIM, SPAN_DIM, SPAN_DIM, SPAN_DIM, D_DIM, 0);

    k_renorm_actors<<<(A_ACT + 255) / 256, 256, 0, stream>>>(actor_mat, actors_n);

    // score_link = actors_n @ small^T   (M=2000, N=512, K=300)
    k_gemm_nt_wmma<<<gemm_blocks(A_ACT, E_SP), 256, 0, stream>>>(
        actors_n, small_, (const float*)nullptr, score_link,
        A_ACT, E_SP, D_DIM, D_DIM, D_DIM, E_SP, 0);

    k_lse<<<A_ACT, 256, 0, stream>>>(score_link, score_1);
}